// GINNet_2688649527833
// MI455X (gfx1250) — compile-verified
//
#include <hip/hip_runtime.h>

// ---------------------------------------------------------------------------
// GIN on MI455X (gfx1250).
// Memory-bound scatter (L2-resident, native f32 atomics) + fp32 WMMA MLPs.
// ---------------------------------------------------------------------------

typedef float v2f __attribute__((ext_vector_type(2)));
typedef float v8f __attribute__((ext_vector_type(8)));

#define HIDC  128      // feature width (IN_C == HID == 128)
#define MTILE 32       // node rows per block
#define PITCH 132      // LDS row pitch in floats: (row*132 + k) % 64 covers all
                       // banks once across the 32 lanes of an A-frag load.

// ---------------------------------------------------------------------------
// Edge scatter-add: agg[dst] += h[src].  One wave (32 lanes) per edge,
// float4 per lane -> one coalesced 512B gather + 4 L2 f32 atomics per lane.
// ---------------------------------------------------------------------------
__global__ __launch_bounds__(256) void gin_scatter_kernel(
    const float* __restrict__ h, const long long* __restrict__ ei,
    float* __restrict__ agg, int ne)
{
    const int edge = (int)((blockIdx.x * 256u + threadIdx.x) >> 5);
    const int lane = threadIdx.x & 31;
    if (edge >= ne) return;
    const int src = (int)ei[edge];
    const int dst = (int)ei[(size_t)ne + edge];
    const float4 v = ((const float4*)(h + (size_t)src * HIDC))[lane];
    float* p = agg + (size_t)dst * HIDC + lane * 4;
    __hip_atomic_fetch_add(p + 0, v.x, __ATOMIC_RELAXED, __HIP_MEMORY_SCOPE_AGENT);
    __hip_atomic_fetch_add(p + 1, v.y, __ATOMIC_RELAXED, __HIP_MEMORY_SCOPE_AGENT);
    __hip_atomic_fetch_add(p + 2, v.z, __ATOMIC_RELAXED, __HIP_MEMORY_SCOPE_AGENT);
    __hip_atomic_fetch_add(p + 3, v.w, __ATOMIC_RELAXED, __HIP_MEMORY_SCOPE_AGENT);
}

// ---------------------------------------------------------------------------
// Fused GIN MLP:  out = (relu_out? relu : id)( relu((x+agg)@Wa + ba) @ Wb + bb )
// Block = 256 threads = 8 waves; 32-node M tile; wave w owns output columns
// [16w, 16w+16); two 16x16 accumulator tiles per wave (row tiles 0 and 1).
// K loop uses V_WMMA_F32_16X16X4_F32 (fp32 A/B, fp32 acc) — full precision.
// ---------------------------------------------------------------------------
__global__ __launch_bounds__(256) void gin_mlp_kernel(
    const float* __restrict__ xin, const float* __restrict__ agg,
    const float* __restrict__ Wa,  const float* __restrict__ ba,
    const float* __restrict__ Wb,  const float* __restrict__ bb,
    float* __restrict__ out, int nrows, int relu_out)
{
    __shared__ float hin [MTILE * PITCH];   // x + agg tile
    __shared__ float hmid[MTILE * PITCH];   // relu((x+agg)@Wa + ba) tile

    const int tid  = threadIdx.x;
    const int wave = tid >> 5;
    const int lane = tid & 31;
    const int half = lane >> 4;      // 0: lanes 0-15, 1: lanes 16-31
    const int r16  = lane & 15;
    const int col  = wave * 16 + r16;          // output column 0..127
    const int base_row = blockIdx.x * MTILE;

    // ---- Stage hin = x + agg (32 rows x 128 cols; zero-fill past N) -------
    {
        const int r    = tid >> 3;             // 0..31
        const int c0   = (tid & 7) * 16;       // 16 floats per thread
        const int grow = base_row + r;
        float* dst = hin + r * PITCH + c0;
        if (grow < nrows) {
            const float4* px = (const float4*)(xin + (size_t)grow * HIDC + c0);
            const float4* pa = (const float4*)(agg + (size_t)grow * HIDC + c0);
            #pragma unroll
            for (int j = 0; j < 4; ++j) {
                float4 a = px[j], b = pa[j];
                dst[4*j + 0] = a.x + b.x;
                dst[4*j + 1] = a.y + b.y;
                dst[4*j + 2] = a.z + b.z;
                dst[4*j + 3] = a.w + b.w;
            }
        } else {
            #pragma unroll
            for (int j = 0; j < 16; ++j) dst[j] = 0.0f;
        }
    }
    __syncthreads();

    // ---- GEMM1: hmid = relu(hin @ Wa + ba) --------------------------------
    {
        v8f acc0 = {}; v8f acc1 = {};
        for (int kk = 0; kk < HIDC; kk += 4) {
            const int kb = kk + 2 * half;      // A/B frag K layout (ISA 7.12.2)
            v2f bfrag;
            bfrag[0] = Wa[(size_t)(kb + 0) * HIDC + col];
            bfrag[1] = Wa[(size_t)(kb + 1) * HIDC + col];
            v2f a0 = *(const v2f*)(hin + (0 * 16 + r16) * PITCH + kb);
            v2f a1 = *(const v2f*)(hin + (1 * 16 + r16) * PITCH + kb);
            acc0 = __builtin_amdgcn_wmma_f32_16x16x4_f32(
                       false, a0, false, bfrag, (short)0, acc0, false, false);
            acc1 = __builtin_amdgcn_wmma_f32_16x16x4_f32(
                       false, a1, false, bfrag, (short)0, acc1, false, false);
        }
        const float bias = ba[col];
        #pragma unroll
        for (int v = 0; v < 8; ++v) {          // C/D layout: M = v + 8*half
            const int row0 = 0 * 16 + v + 8 * half;
            const int row1 = 1 * 16 + v + 8 * half;
            float v0 = acc0[v] + bias; v0 = v0 > 0.0f ? v0 : 0.0f;
            float v1 = acc1[v] + bias; v1 = v1 > 0.0f ? v1 : 0.0f;
            hmid[row0 * PITCH + col] = v0;
            hmid[row1 * PITCH + col] = v1;
        }
    }
    __syncthreads();

    // ---- GEMM2: out = hmid @ Wb + bb (optional relu) ----------------------
    {
        v8f acc0 = {}; v8f acc1 = {};
        for (int kk = 0; kk < HIDC; kk += 4) {
            const int kb = kk + 2 * half;
            v2f bfrag;
            bfrag[0] = Wb[(size_t)(kb + 0) * HIDC + col];
            bfrag[1] = Wb[(size_t)(kb + 1) * HIDC + col];
            v2f a0 = *(const v2f*)(hmid + (0 * 16 + r16) * PITCH + kb);
            v2f a1 = *(const v2f*)(hmid + (1 * 16 + r16) * PITCH + kb);
            acc0 = __builtin_amdgcn_wmma_f32_16x16x4_f32(
                       false, a0, false, bfrag, (short)0, acc0, false, false);
            acc1 = __builtin_amdgcn_wmma_f32_16x16x4_f32(
                       false, a1, false, bfrag, (short)0, acc1, false, false);
        }
        const float bias = bb[col];
        #pragma unroll
        for (int v = 0; v < 8; ++v) {
            const int row0 = 0 * 16 + v + 8 * half;
            const int row1 = 1 * 16 + v + 8 * half;
            float v0 = acc0[v] + bias;
            float v1 = acc1[v] + bias;
            if (relu_out) {
                v0 = v0 > 0.0f ? v0 : 0.0f;
                v1 = v1 > 0.0f ? v1 : 0.0f;
            }
            const int g0 = base_row + row0;
            const int g1 = base_row + row1;
            if (g0 < nrows) out[(size_t)g0 * HIDC + col] = v0;
            if (g1 < nrows) out[(size_t)g1 * HIDC + col] = v1;
        }
    }
}

// ---------------------------------------------------------------------------
// Final FC: out[i, 0:2] = h2[i, :] @ Wfc + bfc.   N*2 outputs, thread/node.
// Wfc (128x2) + bfc stay L0/L2 resident.
// ---------------------------------------------------------------------------
__global__ __launch_bounds__(256) void gin_fc_kernel(
    const float* __restrict__ h, const float* __restrict__ W,
    const float* __restrict__ b, float* __restrict__ out, int nrows)
{
    const int i = blockIdx.x * 256 + threadIdx.x;
    if (i >= nrows) return;
    const float* row = h + (size_t)i * HIDC;
    float acc0 = b[0], acc1 = b[1];
    #pragma unroll 8
    for (int k = 0; k < HIDC; ++k) {
        const float v = row[k];
        acc0 += v * W[2 * k + 0];
        acc1 += v * W[2 * k + 1];
    }
    out[2 * i + 0] = acc0;
    out[2 * i + 1] = acc1;
}

// ---------------------------------------------------------------------------
extern "C" void kernel_launch(void* const* d_in, const int* in_sizes, int n_in,
                              void* d_out, int out_size, void* d_ws, size_t ws_size,
                              hipStream_t stream)
{
    const float*     x   = (const float*)d_in[0];
    const long long* ei  = (const long long*)d_in[1];   // int64 [2, E]
    const float*     W1a = (const float*)d_in[2];
    const float*     b1a = (const float*)d_in[3];
    const float*     W1b = (const float*)d_in[4];
    const float*     b1b = (const float*)d_in[5];
    const float*     W2a = (const float*)d_in[6];
    const float*     b2a = (const float*)d_in[7];
    const float*     W2b = (const float*)d_in[8];
    const float*     b2b = (const float*)d_in[9];
    const float*     Wfc = (const float*)d_in[10];
    const float*     bfc = (const float*)d_in[11];
    float* out = (float*)d_out;

    const int n = in_sizes[0] / HIDC;   // 50000
    const int e = in_sizes[1] / 2;      // 800000

    const size_t featElems = (size_t)n * HIDC;
    float* agg = (float*)d_ws;
    float* h1  = agg + featElems;
    float* h2  = h1  + featElems;
    const size_t featBytes = featElems * sizeof(float);

    const int scatterBlocks = (e + 7) / 8;            // 8 waves/block, wave/edge
    const int mlpBlocks     = (n + MTILE - 1) / MTILE;

    // ---- Layer 1 ----------------------------------------------------------
    hipMemsetAsync(agg, 0, featBytes, stream);
    gin_scatter_kernel<<<scatterBlocks, 256, 0, stream>>>(x, ei, agg, e);
    gin_mlp_kernel<<<mlpBlocks, 256, 0, stream>>>(x, agg, W1a, b1a, W1b, b1b,
                                                  h1, n, /*relu_out=*/1);
    // ---- Layer 2 ----------------------------------------------------------
    hipMemsetAsync(agg, 0, featBytes, stream);
    gin_scatter_kernel<<<scatterBlocks, 256, 0, stream>>>(h1, ei, agg, e);
    gin_mlp_kernel<<<mlpBlocks, 256, 0, stream>>>(h1, agg, W2a, b2a, W2b, b2b,
                                                  h2, n, /*relu_out=*/0);
    // ---- FC ---------------------------------------------------------------
    gin_fc_kernel<<<(n + 255) / 256, 256, 0, stream>>>(h2, Wfc, bfc, out, n);
}